// AttentionEdgeEmb_34256659153219
// MI455X (gfx1250) — compile-verified
//
#include <hip/hip_runtime.h>
#include <hip/hip_bf16.h>

typedef __attribute__((ext_vector_type(2))) float v2f;
typedef __attribute__((ext_vector_type(8))) float v8f;

#define IN_CH 64
#define FEAT  128
#define NEG_BIG (-3.402823466e38f)

// ---------------------------------------------------------------------------
// K1: per-edge score via WMMA f32 16x16x4 (B = ones => WMMA does K-reduction),
//     plus per-block max. Each wave handles 16 edges; block = 8 waves = 128.
// ---------------------------------------------------------------------------
__global__ __launch_bounds__(256) void k_scores(
    const float* __restrict__ emb, const long long* __restrict__ eidx,
    const float* __restrict__ attn_w, const float* __restrict__ attn_b,
    float* __restrict__ scores, float* __restrict__ blk_max, int E)
{
    __shared__ float sW[FEAT];
    __shared__ float sMax[16];
    const int tid = threadIdx.x;
    if (tid < FEAT) sW[tid] = attn_w[tid];
    __syncthreads();

    const int wave  = tid >> 5;     // 0..7
    const int lane  = tid & 31;
    const int m     = lane & 15;    // A-matrix row (edge within wave tile)
    const int khalf = lane >> 4;    // lanes 16..31 hold K=2,3 per ISA A layout

    long long e  = (long long)blockIdx.x * 128 + wave * 16 + m;
    long long ec = (e < E) ? e : (long long)(E - 1);   // clamp, keep EXEC all-1s
    const long long si = eidx[ec];
    const long long ti = eidx[(long long)E + ec];
    const float* __restrict__ sp = emb + si * IN_CH;
    const float* __restrict__ tp = emb + ti * IN_CH;

    v8f acc = {};
    v2f one; one.x = 1.0f; one.y = 1.0f;
    const int cbase = khalf * 2;

    #pragma unroll
    for (int kk = 0; kk < 32; ++kk) {
        const int c0 = kk * 4 + cbase;                 // even, c0 and c0+1 same half
        const float* base = (c0 < IN_CH) ? (sp + c0) : (tp + (c0 - IN_CH));
        v2f a;
        a.x = base[0] * sW[c0];
        a.y = base[1] * sW[c0 + 1];
        // D[m][n] += sum_k A[m][k] * 1  -> every column of D = partial score
        acc = __builtin_amdgcn_wmma_f32_16x16x4_f32(
            /*neg_a=*/false, a, /*neg_b=*/false, one,
            /*c_mod=*/(short)0, acc, /*reuse_a=*/false, /*reuse_b=*/false);
    }

    const float bias = attn_b[0];
    float lmax = NEG_BIG;
    if (m == 0) {
        // lane 0: M=0..7 in acc[0..7]; lane 16: M=8..15 (ISA C/D layout, N=0)
        const long long ebase = (long long)blockIdx.x * 128 + wave * 16 + khalf * 8;
        #pragma unroll
        for (int r = 0; r < 8; ++r) {
            const float s = acc[r] + bias;
            const long long ee = ebase + r;
            if (ee < E) { scores[ee] = s; lmax = fmaxf(lmax, s); }
        }
        sMax[wave * 2 + khalf] = lmax;
    }
    __syncthreads();
    if (tid == 0) {
        float bm = sMax[0];
        #pragma unroll
        for (int i = 1; i < 16; ++i) bm = fmaxf(bm, sMax[i]);
        blk_max[blockIdx.x] = bm;
    }
}

// ---------------------------------------------------------------------------
// K2: single-block deterministic max reduction of block maxima
// ---------------------------------------------------------------------------
__global__ __launch_bounds__(256) void k_reduce_max(
    const float* __restrict__ in, int n, float* __restrict__ out)
{
    __shared__ float sm[256];
    float m = NEG_BIG;
    for (int i = threadIdx.x; i < n; i += 256) m = fmaxf(m, in[i]);
    sm[threadIdx.x] = m;
    __syncthreads();
    for (int s = 128; s > 0; s >>= 1) {
        if (threadIdx.x < (unsigned)s) sm[threadIdx.x] = fmaxf(sm[threadIdx.x], sm[threadIdx.x + s]);
        __syncthreads();
    }
    if (threadIdx.x == 0) out[0] = sm[0];
}

// ---------------------------------------------------------------------------
// K3: per-block partial sums of exp(score - gmax), fixed grid-stride order
// ---------------------------------------------------------------------------
__global__ __launch_bounds__(256) void k_expsum(
    const float* __restrict__ scores, int E, const float* __restrict__ gmax,
    float* __restrict__ blk_sum)
{
    __shared__ float sm[256];
    const float g = gmax[0];
    float acc = 0.0f;
    const long long stride = (long long)gridDim.x * 256;
    for (long long i = (long long)blockIdx.x * 256 + threadIdx.x; i < E; i += stride)
        acc += expf(scores[i] - g);
    sm[threadIdx.x] = acc;
    __syncthreads();
    for (int s = 128; s > 0; s >>= 1) {
        if (threadIdx.x < (unsigned)s) sm[threadIdx.x] += sm[threadIdx.x + s];
        __syncthreads();
    }
    if (threadIdx.x == 0) blk_sum[blockIdx.x] = sm[0];
}

// ---------------------------------------------------------------------------
// K4: single-block deterministic sum reduction -> reciprocal
// ---------------------------------------------------------------------------
__global__ __launch_bounds__(256) void k_reduce_sum(
    const float* __restrict__ in, int n, float* __restrict__ out)
{
    __shared__ float sm[256];
    float a = 0.0f;
    for (int i = threadIdx.x; i < n; i += 256) a += in[i];
    sm[threadIdx.x] = a;
    __syncthreads();
    for (int s = 128; s > 0; s >>= 1) {
        if (threadIdx.x < (unsigned)s) sm[threadIdx.x] += sm[threadIdx.x + s];
        __syncthreads();
    }
    if (threadIdx.x == 0) out[0] = 1.0f / sm[0];
}

// ---------------------------------------------------------------------------
// K5: out[e, :] = attn_e * concat(src_row, tgt_row).
//     One wave per edge: lanes 0-15 stream src row, 16-31 tgt row, float4 I/O.
//     Writes 512B contiguous per wave -> saturates the dominant HBM-write cost.
// ---------------------------------------------------------------------------
__global__ __launch_bounds__(256) void k_out(
    const float* __restrict__ emb, const long long* __restrict__ eidx,
    const float* __restrict__ scores, const float* __restrict__ gmax,
    const float* __restrict__ invs, float* __restrict__ out, int E)
{
    const int wave = threadIdx.x >> 5;
    const int lane = threadIdx.x & 31;
    const long long e = (long long)blockIdx.x * 8 + wave;
    if (e >= E) return;

    const float attn = expf(scores[e] - gmax[0]) * invs[0];
    const long long row = (lane < 16) ? eidx[e] : eidx[(long long)E + e];
    const int csub = 4 * (lane & 15);

    const float4 v = *(const float4*)(emb + row * IN_CH + csub);
    float4 o;
    o.x = v.x * attn; o.y = v.y * attn; o.z = v.z * attn; o.w = v.w * attn;
    *(float4*)(out + e * (long long)FEAT + (lane >> 4) * IN_CH + csub) = o;
}

extern "C" void kernel_launch(void* const* d_in, const int* in_sizes, int n_in,
                              void* d_out, int out_size, void* d_ws, size_t ws_size,
                              hipStream_t stream) {
    const float*     emb    = (const float*)d_in[0];
    const long long* eidx   = (const long long*)d_in[1];   // int64 [2, E]
    const float*     attn_w = (const float*)d_in[2];       // [1, 128]
    const float*     attn_b = (const float*)d_in[3];       // [1]
    float*           out    = (float*)d_out;

    const int E   = in_sizes[1] / 2;
    const int nb1 = (E + 127) / 128;
    const int NB3 = 2048;

    // workspace layout (floats): scores[E] | blk_max[nb1] | gmax | blk_sum[NB3] | inv_sum
    float* scores  = (float*)d_ws;
    float* blk_max = scores + E;
    float* gmax    = blk_max + nb1;
    float* blk_sum = gmax + 1;
    float* invs    = blk_sum + NB3;

    k_scores    <<<nb1,         256, 0, stream>>>(emb, eidx, attn_w, attn_b, scores, blk_max, E);
    k_reduce_max<<<1,           256, 0, stream>>>(blk_max, nb1, gmax);
    k_expsum    <<<NB3,         256, 0, stream>>>(scores, E, gmax, blk_sum);
    k_reduce_sum<<<1,           256, 0, stream>>>(blk_sum, NB3, invs);
    k_out       <<<(E + 7) / 8, 256, 0, stream>>>(emb, eidx, scores, gmax, invs, out, E);
}